// HighOrderActivation_69140383531252
// MI455X (gfx1250) — compile-verified
//
#include <hip/hip_runtime.h>

// High-order activation: out[b,g,:] = sum_k coef[b,g,k] * params[g, idx[b,g,k], :]
// Reformulated as Out[16x64] = W[16x16] x P[16x64] per (group, 16-batch tile),
// executed with v_wmma_f32_16x16x4_f32 (full fp32; problem is HBM-write bound,
// ~142MB traffic -> ~6us floor at 23.3 TB/s, so fp32 WMMA is the right choice).

#define BATCH   1024
#define NGROUP  512
#define ARITY   4
#define ODIM    64
#define TABLE   16
#define LDS_STRIDE 80  // 64 + 16 pad: half-wave LDS bank sets disjoint (64 banks)

typedef __attribute__((ext_vector_type(2))) float v2f;
typedef __attribute__((ext_vector_type(8))) float v8f;

#if __has_builtin(__builtin_amdgcn_wmma_f32_16x16x4_f32)
#define HAVE_WMMA_F32X4 1
#else
#define HAVE_WMMA_F32X4 0
#endif

// branchless compare-swap keeping original indices (EXEC stays all-ones)
#define CSWAP(a, b, ia, ib)                              \
  {                                                      \
    float ta = (a), tb = (b);                            \
    int qa = (ia), qb = (ib);                            \
    bool sw = ta > tb;                                   \
    (a) = sw ? tb : ta;                                  \
    (b) = sw ? ta : tb;                                  \
    (ia) = sw ? qb : qa;                                 \
    (ib) = sw ? qa : qb;                                 \
  }

__global__ __launch_bounds__(128) void hoact_wmma_kernel(
    const float* __restrict__ X,   // [BATCH, NGROUP, ARITY]
    const float* __restrict__ P,   // [NGROUP, TABLE, ODIM]
    float* __restrict__ O) {       // [BATCH, NGROUP, ODIM]
  __shared__ float lds[TABLE * LDS_STRIDE];

  const int g = blockIdx.y;
  const int tid = threadIdx.x;

  // ---- stage params[g] (16x64 fp32 = 4KB) into padded LDS ----
  {
    const float* pg = P + (size_t)g * (TABLE * ODIM);
    for (int t = tid; t < TABLE * ODIM; t += 128) {
      lds[(t >> 6) * LDS_STRIDE + (t & 63)] = pg[t];
    }
  }
  __syncthreads();

  const int wave = tid >> 5;
  const int lane = tid & 31;
  const int row  = lane & 15;       // M for A-frag, N for B-frag, per ISA layout
  const int hi   = lane >> 4;       // 0 -> K/M low half, 1 -> high half
  const int bbase = blockIdx.x * 64 + wave * 16;
  const int b = bbase + row;

  // ---- per-batch-row: sort 4 values, barycentric coefs, vertex indices ----
  const float4 xv = *(const float4*)(X + ((size_t)b * NGROUP + g) * ARITY);
  float v0 = xv.x, v1 = xv.y, v2 = xv.z, v3 = xv.w;
  int i0 = 0, i1 = 1, i2 = 2, i3 = 3;
  CSWAP(v0, v1, i0, i1)
  CSWAP(v2, v3, i2, i3)
  CSWAP(v0, v2, i0, i2)
  CSWAP(v1, v3, i1, i3)
  CSWAP(v1, v2, i1, i2)
  // coef = [min, successive gaps]; idx = reversed cumsum of 2^argsort
  const float c0 = v0, c1 = v1 - v0, c2 = v2 - v1, c3 = v3 - v2;
  const int id3 = 1 << i3;
  const int id2 = id3 + (1 << i2);
  const int id1 = id2 + (1 << i1);
  const int id0 = 15;  // full sum of distinct powers of two

  // sparse 16-wide weight row: at most one idx matches any vertex Kv
  auto wv = [&](int Kv) -> float {
    float w = (id0 == Kv) ? c0 : 0.0f;
    w += (id1 == Kv) ? c1 : 0.0f;
    w += (id2 == Kv) ? c2 : 0.0f;
    w += (id3 == Kv) ? c3 : 0.0f;
    return w;
  };

#if HAVE_WMMA_F32X4
  // A fragments for the four K-chunks (16x4 fp32 layout: lane holds M=row,
  // VGPR v holds K = 4*c + 2*hi + v)
  v2f a0, a1, a2, a3;
  a0[0] = wv(0 + 2 * hi);  a0[1] = wv(0 + 2 * hi + 1);
  a1[0] = wv(4 + 2 * hi);  a1[1] = wv(4 + 2 * hi + 1);
  a2[0] = wv(8 + 2 * hi);  a2[1] = wv(8 + 2 * hi + 1);
  a3[0] = wv(12 + 2 * hi); a3[1] = wv(12 + 2 * hi + 1);

#pragma unroll
  for (int nt = 0; nt < 4; ++nt) {  // N=64 in 4 tiles of 16
    v8f acc = {};
#pragma unroll
    for (int c = 0; c < 4; ++c) {   // K=16 in 4 chunks of 4
      const int Kb = 4 * c + 2 * hi;
      v2f bfrag;
      bfrag[0] = lds[Kb * LDS_STRIDE + nt * 16 + row];
      bfrag[1] = lds[(Kb + 1) * LDS_STRIDE + nt * 16 + row];
      const v2f afrag = (c == 0) ? a0 : (c == 1) ? a1 : (c == 2) ? a2 : a3;
      acc = __builtin_amdgcn_wmma_f32_16x16x4_f32(
          /*neg_a=*/false, afrag, /*neg_b=*/false, bfrag,
          /*c_mod=*/(short)0, acc, /*reuse_a=*/false, /*reuse_b=*/false);
    }
    // D layout: VGPR r -> M = r + 8*hi, N = nt*16 + row; 64B-contiguous per half-wave
#pragma unroll
    for (int r = 0; r < 8; ++r) {
      O[(((size_t)(bbase + r + 8 * hi)) * NGROUP + g) * ODIM + nt * 16 + row] = acc[r];
    }
  }
#else
  // Fallback (no f32 16x16x4 WMMA builtin): direct 4-row gather weighted sum.
  if (hi == 0) {
    float* op = O + ((size_t)b * NGROUP + g) * ODIM;
    const float* r0 = lds + id0 * LDS_STRIDE;
    const float* r1 = lds + id1 * LDS_STRIDE;
    const float* r2 = lds + id2 * LDS_STRIDE;
    const float* r3 = lds + id3 * LDS_STRIDE;
#pragma unroll 8
    for (int d = 0; d < ODIM; ++d) {
      op[d] = c0 * r0[d] + c1 * r1[d] + c2 * r2[d] + c3 * r3[d];
    }
  }
#endif
}

extern "C" void kernel_launch(void* const* d_in, const int* in_sizes, int n_in,
                              void* d_out, int out_size, void* d_ws, size_t ws_size,
                              hipStream_t stream) {
  const float* X = (const float*)d_in[0];   // [1024, 512, 4] fp32
  const float* P = (const float*)d_in[1];   // [512, 16, 64] fp32
  float* O = (float*)d_out;                 // [1024, 512, 64] fp32

  dim3 grid(BATCH / 64, NGROUP);            // 16 x 512 blocks
  dim3 block(128);                          // 4 waves; wave handles a 16-batch tile
  hipLaunchKernelGGL(hoact_wmma_kernel, grid, block, 0, stream, X, P, O);
}